// MultiHeadAttModel_allin_52536039964906
// MI455X (gfx1250) — compile-verified
//
#include <hip/hip_runtime.h>
#include <hip/hip_bf16.h>
#include <math.h>

// ---------------------------------------------------------------------------
// Problem constants (from reference):
//   B=16, NA=64, D=128, N=4*NA=256, NV=4 heads, DV=32
// Refactor: relu((kf+pos)W^T + b) = relu((kf W^T + b) + adjs_pos (pe W^T))
// so we never materialize the 134MB nb tensor.
// ---------------------------------------------------------------------------

#define BB   16
#define NA_  64
#define DD   128
#define NN   256   // 4*NA
#define NVH  4
#define DVH  32

typedef __attribute__((ext_vector_type(2))) float v2f;
typedef __attribute__((ext_vector_type(8))) float v8f;

// ---------------------------------------------------------------------------
// WMMA f32 GEMM:  C[M,N] = act(A[M,K] @ W[N,K]^T + bias[N])
// K, N multiples of 16 (here K=N=128). One wave per 16x16 C tile.
// Fragment layouts per CDNA5 ISA 7.12.2:
//   A 16x4 f32: lanes 0-15 -> rows, vgpr0=K0 vgpr1=K1; lanes 16-31 -> K2,K3
//   B 4x16 f32: mirrored with N in place of M (B = W^T, row-major W[N,K])
//   C 16x16 f32: vgpr r -> row r (lanes 0-15) / row r+8 (lanes 16-31), col=lane&15
// ---------------------------------------------------------------------------
__global__ __launch_bounds__(128)
void wmma_gemm_nt_f32(const float* __restrict__ A,
                      const float* __restrict__ W,
                      const float* __restrict__ bias,
                      float* __restrict__ C,
                      int M, int N, int K, int relu)
{
  const int lane  = threadIdx.x & 31;
  const int wave  = threadIdx.x >> 5;
  const int ntile = blockIdx.x * 4 + wave;
  const int mtile = blockIdx.y;
  if (ntile * 16 >= N || mtile * 16 >= M) return;

  const int m0   = mtile * 16;
  const int n0   = ntile * 16;
  const int half = lane >> 4;   // 0 -> K0/K1, 1 -> K2/K3
  const int lid  = lane & 15;

  const float* __restrict__ arow = A + (size_t)(m0 + lid) * K + half * 2;
  const float* __restrict__ wrow = W + (size_t)(n0 + lid) * K + half * 2;

  __builtin_prefetch(wrow, 0, 3);   // global_prefetch_b8 on the weight tile

  v8f acc = {};
  #pragma unroll
  for (int k = 0; k < 128; k += 4) {
    v2f a = *(const v2f*)(arow + k);
    v2f b = *(const v2f*)(wrow + k);
    // D = A x B + C  :  v_wmma_f32_16x16x4_f32
    acc = __builtin_amdgcn_wmma_f32_16x16x4_f32(
        /*neg_a=*/false, a, /*neg_b=*/false, b,
        /*c_mod=*/(short)0, acc, /*reuse_a=*/false, /*reuse_b=*/false);
  }

  const float bv = bias[n0 + lid];   // column of this lane (same for all 8 rows)
  #pragma unroll
  for (int r = 0; r < 8; ++r) {
    float v = acc[r] + bv;
    if (relu) v = fmaxf(v, 0.0f);
    C[(size_t)(m0 + r + half * 8) * N + (n0 + lid)] = v;
  }
}

// ---------------------------------------------------------------------------
// peW[p,d] = sum_e pe[p,e] * W[d,e]   (pe computed analytically, p<6, d<128)
// pe[p,2i] = sin(p*div_i), pe[p,2i+1] = cos(p*div_i), div_i = exp(-2i*ln(1e4)/128)
// ---------------------------------------------------------------------------
__global__ __launch_bounds__(128)
void pe_proj_kernel(const float* __restrict__ Wk, const float* __restrict__ Wv,
                    float* __restrict__ peWk, float* __restrict__ peWv)
{
  const int p = blockIdx.x;     // 0..5
  const int d = threadIdx.x;    // 0..127
  const float kdiv = -9.210340371976184f / 128.0f;   // -ln(10000)/128
  float sk = 0.f, sv = 0.f;
  #pragma unroll 8
  for (int e = 0; e < 128; ++e) {
    const int  i   = e & ~1;                 // even index 2i
    const float dv = __expf((float)i * kdiv);
    const float ang = (float)p * dv;
    const float pe = (e & 1) ? __cosf(ang) : __sinf(ang);
    sk += pe * Wk[d * 128 + e];
    sv += pe * Wv[d * 128 + e];
  }
  peWk[p * 128 + d] = sk;
  peWv[p * 128 + d] = sv;
}

// ---------------------------------------------------------------------------
// Pk/Pv[a,n,d] = sum_{p<6} adjs_pos[a,n,p] * peW{k,v}[p,d]
// ---------------------------------------------------------------------------
__global__ __launch_bounds__(256)
void pos_proj_kernel(const float* __restrict__ adjs_pos,
                     const float* __restrict__ peWk,
                     const float* __restrict__ peWv,
                     float* __restrict__ Pk, float* __restrict__ Pv)
{
  const size_t idx = (size_t)blockIdx.x * 256 + threadIdx.x;  // over NA*N*D
  if (idx >= (size_t)NA_ * NN * DD) return;
  const int    d  = (int)(idx & (DD - 1));
  const size_t an = idx >> 7;
  const float* __restrict__ ap = adjs_pos + an * 6;
  float sk = 0.f, sv = 0.f;
  #pragma unroll
  for (int p = 0; p < 6; ++p) {
    const float c = ap[p];
    sk += c * peWk[p * 128 + d];
    sv += c * peWv[p * 128 + d];
  }
  Pk[idx] = sk;
  Pv[idx] = sv;
}

// ---------------------------------------------------------------------------
// Fused attention: one block per (b,a).
//   kh[n,:]  = relu(Kp[b,n,:] + Pk[a,n,:])          (bias folded into Kp)
//   s[h,n]   = qh[b,a,h*32:..] . kh[n,h*32:..] / sqrt(128), masked
//   att      = softmax_n(s)
//   out[h,d] = sum_n att[h,n] * relu(Vp[b,n,hd] + Pv[a,n,hd])
// ---------------------------------------------------------------------------
__global__ __launch_bounds__(256)
void attn_kernel(const float* __restrict__ qh,
                 const float* __restrict__ Kp,
                 const float* __restrict__ Vp,
                 const float* __restrict__ Pk,
                 const float* __restrict__ Pv,
                 const int*   __restrict__ adjs_pre,
                 float* __restrict__ out)
{
  __shared__ float q_sh[DD];
  __shared__ float sc[NVH][NN];
  __shared__ float ssum[NVH];

  const int a = blockIdx.x;
  const int b = blockIdx.y;
  const int t = threadIdx.x;      // 0..255

  if (t < DD) q_sh[t] = qh[((size_t)b * NA_ + a) * DD + t];
  __syncthreads();

  // ---- scores: thread t owns neighbor n = t -------------------------------
  {
    const float* __restrict__ kpr = Kp + ((size_t)b * NN + t) * DD;
    const float* __restrict__ pkr = Pk + ((size_t)a * NN + t) * DD;
    const float inv_sqrtD = 0.08838834764831845f;           // 1/sqrt(128)
    const int   m = adjs_pre[a * NA_ + (t & (NA_ - 1))];    // tile(adjs_pre,(1,4))
    #pragma unroll
    for (int h = 0; h < NVH; ++h) {
      float s = 0.f;
      #pragma unroll
      for (int d = 0; d < DVH; ++d) {
        const int hd = h * DVH + d;
        const float kv = fmaxf(kpr[hd] + pkr[hd], 0.0f);
        s += q_sh[hd] * kv;
      }
      sc[h][t] = (m == 0) ? -1e9f : s * inv_sqrtD;
    }
  }
  __syncthreads();

  // ---- softmax over n: wave h (h<4) reduces head h ------------------------
  {
    const int w    = t >> 5;
    const int lane = t & 31;
    if (w < NVH) {
      float mx = -1e30f;
      for (int j = lane; j < NN; j += 32) mx = fmaxf(mx, sc[w][j]);
      #pragma unroll
      for (int off = 16; off > 0; off >>= 1) mx = fmaxf(mx, __shfl_xor(mx, off, 32));
      float sum = 0.f;
      for (int j = lane; j < NN; j += 32) {
        const float e = __expf(sc[w][j] - mx);
        sc[w][j] = e;
        sum += e;
      }
      #pragma unroll
      for (int off = 16; off > 0; off >>= 1) sum += __shfl_xor(sum, off, 32);
      if (lane == 0) ssum[w] = sum;
    }
  }
  __syncthreads();

  // ---- output: thread t (<128) owns out dim t -----------------------------
  if (t < DD) {
    const int h = t >> 5;
    const float* __restrict__ vpr = Vp + (size_t)b * NN * DD + t;
    const float* __restrict__ pvr = Pv + (size_t)a * NN * DD + t;
    float acc = 0.f;
    for (int n = 0; n < NN; ++n) {
      const float vv = fmaxf(vpr[(size_t)n * DD] + pvr[(size_t)n * DD], 0.0f);
      acc += sc[h][n] * vv;
    }
    out[((size_t)b * NA_ + a) * DD + t] = acc / ssum[h];
  }
}

// ---------------------------------------------------------------------------
// Host-side orchestration.
// Inputs: 0:q 1:k 2:adjs_pos 3:Wq 4:bq 5:Wk 6:bk 7:Wv 8:bv 9:Wo 10:bo 11:adjs_pre
// ---------------------------------------------------------------------------
extern "C" void kernel_launch(void* const* d_in, const int* in_sizes, int n_in,
                              void* d_out, int out_size, void* d_ws, size_t ws_size,
                              hipStream_t stream)
{
  const float* q        = (const float*)d_in[0];
  const float* kf       = (const float*)d_in[1];   // [B,NA,4,D] == [B,N,D] flat
  const float* adjs_pos = (const float*)d_in[2];
  const float* Wq       = (const float*)d_in[3];
  const float* bq       = (const float*)d_in[4];
  const float* Wk       = (const float*)d_in[5];
  const float* bk       = (const float*)d_in[6];
  const float* Wv       = (const float*)d_in[7];
  const float* bv       = (const float*)d_in[8];
  const float* Wo       = (const float*)d_in[9];
  const float* bo       = (const float*)d_in[10];
  const int*   adjs_pre = (const int*)d_in[11];

  float* ws   = (float*)d_ws;
  float* peWk = ws;                          // 6*128
  float* peWv = peWk + 6 * 128;              // 6*128
  float* qh   = peWv + 6 * 128;              // B*NA*D      = 131072
  float* Kp   = qh   + (size_t)BB * NA_ * DD; // B*N*D      = 524288
  float* Vp   = Kp   + (size_t)BB * NN  * DD; // B*N*D      = 524288
  float* Pk   = Vp   + (size_t)BB * NN  * DD; // NA*N*D     = 2097152
  float* Pv   = Pk   + (size_t)NA_ * NN * DD; // NA*N*D     = 2097152
  float* attn = Pv   + (size_t)NA_ * NN * DD; // B*NA*D     = 131072

  // 1) fold sinusoidal PE through Wk/Wv  (peW = pe @ W^T, [6,128])
  pe_proj_kernel<<<6, 128, 0, stream>>>(Wk, Wv, peWk, peWv);

  // 2) dense projections via WMMA f32 (grid.x*4 waves cover N/16 = 8 ntiles)
  wmma_gemm_nt_f32<<<dim3(2, (BB * NA_) / 16), 128, 0, stream>>>(
      q, Wq, bq, qh, BB * NA_, DD, DD, /*relu=*/1);
  wmma_gemm_nt_f32<<<dim3(2, (BB * NN) / 16), 128, 0, stream>>>(
      kf, Wk, bk, Kp, BB * NN, DD, DD, /*relu=*/0);
  wmma_gemm_nt_f32<<<dim3(2, (BB * NN) / 16), 128, 0, stream>>>(
      kf, Wv, bv, Vp, BB * NN, DD, DD, /*relu=*/0);

  // 3) positional projections (K=6 contraction)
  pos_proj_kernel<<<(NA_ * NN * DD) / 256, 256, 0, stream>>>(
      adjs_pos, peWk, peWv, Pk, Pv);

  // 4) fused masked softmax attention, one block per (b, a)
  attn_kernel<<<dim3(NA_, BB), 256, 0, stream>>>(
      qh, Kp, Vp, Pk, Pv, adjs_pre, attn);

  // 5) output projection + relu straight into d_out
  wmma_gemm_nt_f32<<<dim3(2, (BB * NA_) / 16), 128, 0, stream>>>(
      attn, Wo, bo, (float*)d_out, BB * NA_, DD, DD, /*relu=*/1);
}